// VQ_72318659330153
// MI455X (gfx1250) — compile-verified
//
#include <hip/hip_runtime.h>

typedef __attribute__((ext_vector_type(2))) float v2f;
typedef __attribute__((ext_vector_type(8))) float v8f;

#define K_CODES 512
#define DIMS    64
#define HW      1024        // 32*32
#define NTILES_K 32         // 512/16

// workspace layout (in floats)
#define WS_WT   0           // 64*512: transposed weight, d-pair interleaved float2
#define WS_WSQ  32768       // 512 squared norms
#define WS_PART 33280       // 512 per-block loss partials

// ---------------- zero out the whole output buffer (covers one-hot zeros) ---
__global__ void vq_zero(float* __restrict__ p, long n) {
    long i = (long)blockIdx.x * blockDim.x + threadIdx.x;
    long stride = (long)gridDim.x * blockDim.x;
    long n4 = n >> 2;
    float4* p4 = (float4*)p;
    for (long j = i; j < n4; j += stride) p4[j] = make_float4(0.f, 0.f, 0.f, 0.f);
    for (long j = (n4 << 2) + i; j < n; j += stride) p[j] = 0.f;
}

// ------- transpose weight -> Wt2[d/2][k] = (w[k][d], w[k][d+1]); ||w_k||^2 --
__global__ void vq_prep(const float* __restrict__ w, float* __restrict__ ws) {
    int n = blockIdx.x * blockDim.x + threadIdx.x;
    if (n >= K_CODES) return;
    const float* row = w + n * DIMS;
    v2f* wt2 = (v2f*)(ws + WS_WT);
    float s = 0.f;
#pragma unroll
    for (int dp = 0; dp < DIMS / 2; ++dp) {
        float v0 = row[2 * dp];
        float v1 = row[2 * dp + 1];
        s += v0 * v0 + v1 * v1;
        v2f pk; pk.x = v0; pk.y = v1;
        wt2[dp * K_CODES + n] = pk;
    }
    ws[WS_WSQ + n] = s;
}

// ---------------- main: distances via FP32 WMMA, argmin, outputs ------------
__global__ __launch_bounds__(256) void vq_main(
        const float* __restrict__ x, const float* __restrict__ w,
        const float* __restrict__ ws, float* __restrict__ out_q,
        float* __restrict__ out_enc, float* __restrict__ partials) {
    __shared__ int   idxLDS[8 * 16];
    __shared__ float psum[8];

    const int lane = threadIdx.x & 31;
    const int wave = threadIdx.x >> 5;
    const int tile = blockIdx.x * 8 + wave;          // 0..4095, 16 pixels each
    const int col  = lane & 15;
    const int half = lane >> 4;

    const int bb  = tile >> 6;                       // batch index (1024 hw / batch)
    const int hw0 = (tile & 63) * 16;                // base h*w within batch
    const float* xb = x + (long)bb * (DIMS * HW);

    // ---- load A tile: 16 rows x 64 dims, ISA 16x4 f32 A layout per chunk ----
    v2f a[16];
    float xsqp = 0.f;
    const int myhw = hw0 + col;
#pragma unroll
    for (int k = 0; k < 16; ++k) {
        int d0 = 4 * k + 2 * half;
        float a0 = xb[d0 * HW + myhw];
        float a1 = xb[(d0 + 1) * HW + myhw];
        a[k].x = a0; a[k].y = a1;
        xsqp += a0 * a0 + a1 * a1;
    }
    float xsq_full = xsqp + __shfl_xor(xsqp, 16, 32);    // lane m holds ||x_m||^2
    // xq[r] = ||x||^2 of C/D-layout row (r + 8*half)
    float xq[8];
#pragma unroll
    for (int r = 0; r < 8; ++r)
        xq[r] = __shfl(xsq_full, r + 8 * half, 32);

    const v2f*   wt2 = (const v2f*)(ws + WS_WT);
    const float* wsq = ws + WS_WSQ;

    float best[8];
    int   bestn[8];
#pragma unroll
    for (int r = 0; r < 8; ++r) { best[r] = 3.4e38f; bestn[r] = 0; }

    for (int t = 0; t < NTILES_K; ++t) {
        float wsqv = wsq[t * 16 + col];
        v8f acc = {};
#pragma unroll
        for (int k = 0; k < 16; ++k) {
            // B fragment: rows d0=4k+2*half, d0+1 at column (t*16+col)
            v2f b = wt2[(2 * k + half) * K_CODES + t * 16 + col];   // one b64 load
            acc = __builtin_amdgcn_wmma_f32_16x16x4_f32(
                    false, a[k], false, b, (short)0, acc, false, false);
        }
        int n = t * 16 + col;
#pragma unroll
        for (int r = 0; r < 8; ++r) {
            float dist = xq[r] + wsqv - 2.0f * acc[r];
            if (dist < best[r]) { best[r] = dist; bestn[r] = n; }
        }
    }

    // ---- argmin across the 16 columns of each half (rows r + 8*half) -------
#pragma unroll
    for (int r = 0; r < 8; ++r) {
#pragma unroll
        for (int mask = 1; mask <= 8; mask <<= 1) {
            float od = __shfl_xor(best[r],  mask, 16);
            int   oi = __shfl_xor(bestn[r], mask, 16);
            if (od < best[r] || (od == best[r] && oi < bestn[r])) {
                best[r] = od; bestn[r] = oi;
            }
        }
        if (col == 0) idxLDS[wave * 16 + 8 * half + r] = bestn[r];
    }
    __syncthreads();

    // ---- per-row outputs. Lanes cover hw (coalesced stores/loads), loop d. --
    // my row (pixel) = col, its codebook index:
    const int myidx = idxLDS[wave * 16 + col];
    float lsum = 0.f;
#pragma unroll
    for (int dd = 0; dd < 32; ++dd) {
        int d = 2 * dd + half;                        // lanes 0-15: even d, 16-31: odd d
        float q  = w[myidx * DIMS + d];               // gather, L2-resident codebook
        float xv = xb[d * HW + myhw];                 // coalesced
        out_q[((long)bb * DIMS + d) * HW + myhw] = q; // coalesced 64B per half-wave
        float e = q - xv;
        lsum += e * e;
    }
    if (half == 0) {                                  // one-hot scatter, 16 stores/wave
        long n = (long)tile * 16 + col;
        out_enc[n * K_CODES + myidx] = 1.0f;
    }
#pragma unroll
    for (int mask = 16; mask >= 1; mask >>= 1) lsum += __shfl_xor(lsum, mask, 32);
    if (lane == 0) psum[wave] = lsum;
    __syncthreads();
    if (threadIdx.x == 0) {
        float s = 0.f;
        for (int i = 0; i < 8; ++i) s += psum[i];     // fixed order: deterministic
        partials[blockIdx.x] = s;
    }
}

// ---------------- deterministic final reduction -----------------------------
__global__ void vq_final(const float* __restrict__ partials, float* __restrict__ loss) {
    if (threadIdx.x == 0 && blockIdx.x == 0) {
        float s = 0.f;
        for (int i = 0; i < 512; ++i) s += partials[i];
        // z_q_loss + 0.25*z_e_loss, both equal mean((q-x)^2) over 64*64*32*32 elems
        loss[0] = 1.25f * s / 4194304.0f;
    }
}

extern "C" void kernel_launch(void* const* d_in, const int* in_sizes, int n_in,
                              void* d_out, int out_size, void* d_ws, size_t ws_size,
                              hipStream_t stream) {
    const float* x = (const float*)d_in[0];     // [64,64,32,32]
    const float* w = (const float*)d_in[1];     // [512,64]
    float* out      = (float*)d_out;
    float* ws       = (float*)d_ws;
    float* out_loss = out;
    float* out_q    = out + 1;
    float* out_enc  = out + 1 + 64 * 64 * 32 * 32;

    vq_zero <<<2048, 256, 0, stream>>>(out, (long)out_size);
    vq_prep <<<2,    256, 0, stream>>>(w, ws);
    vq_main <<<512,  256, 0, stream>>>(x, w, ws, out_q, out_enc, ws + WS_PART);
    vq_final<<<1,    32,  0, stream>>>(ws + WS_PART, out_loss);
}